// SAGE_28432683499968
// MI455X (gfx1250) — compile-verified
//
#include <hip/hip_runtime.h>

#define NNODES 50000
#define HDIM   160
#define HB     32
#define LMDIM  768
#define NTILES 3125            // 50000 / 16
#define NTG    625             // 3125 / 5  (5 row-tiles per wave)

typedef float v2f __attribute__((ext_vector_type(2)));
typedef float v8f __attribute__((ext_vector_type(8)));

__device__ __forceinline__ v8f wmma_f32(v2f a, v2f b, v8f c) {
    // V_WMMA_F32_16X16X4_F32 : D = A(16x4) * B(4x16) + C(16x16), all fp32
    return __builtin_amdgcn_wmma_f32_16x16x4_f32(
        false, a, false, b, (short)0, c, false, false);
}

__device__ __forceinline__ float leaky(float v) { return v > 0.0f ? v : 0.01f * v; }

// ---------------------------------------------------------------------------
// Weight transpose: out[c*K + k] = in[k*C + c]   (in: [K,C] row-major)
// ---------------------------------------------------------------------------
__global__ void transpose_kernel(const float* __restrict__ in, float* __restrict__ out,
                                 int K, int C)
{
    const int tid = blockIdx.x * blockDim.x + threadIdx.x;
    if (tid >= K * C) return;
    const int k = tid / C;
    const int c = tid - k * C;
    out[c * K + k] = in[tid];
}

// ---------------------------------------------------------------------------
// Small projections (K=6 and K=11) -> h0 cols 0..63, plain FMA
// ---------------------------------------------------------------------------
__global__ void proj_small_kernel(const float* __restrict__ num_prop,
                                  const float* __restrict__ num_cat,
                                  const float* __restrict__ W_prop, const float* __restrict__ b_prop,
                                  const float* __restrict__ W_cat,  const float* __restrict__ b_cat,
                                  float* __restrict__ h0)
{
    const int tid = blockIdx.x * blockDim.x + threadIdx.x;
    if (tid >= NNODES * 64) return;
    const int row = tid >> 6;
    const int col = tid & 63;
    float s;
    if (col < 32) {
        s = b_prop[col];
        const float* a = num_prop + (size_t)row * 6;
        #pragma unroll
        for (int k = 0; k < 6; ++k) s += a[k] * W_prop[k * HB + col];
    } else {
        const int c = col - 32;
        s = b_cat[c];
        const float* a = num_cat + (size_t)row * 11;
        #pragma unroll
        for (int k = 0; k < 11; ++k) s += a[k] * W_cat[k * HB + c];
    }
    h0[(size_t)row * HDIM + col] = leaky(s);
}

// ---------------------------------------------------------------------------
// Big projections (K=768) -> h0 cols 64..159.  grid (625, 3), block 32.
// One wave: 5 row-tiles (80 rows) x 32 cols, WT = transposed weight [32,768].
// ---------------------------------------------------------------------------
__global__ __launch_bounds__(32) void proj_big_kernel(
    const float* __restrict__ des, const float* __restrict__ tweet,
    const float* __restrict__ pre_x,
    const float* __restrict__ WT_des,   const float* __restrict__ b_des,
    const float* __restrict__ WT_text,  const float* __restrict__ b_text,
    const float* __restrict__ WT_tweet, const float* __restrict__ b_tweet,
    float* __restrict__ h0)
{
    const int tg   = blockIdx.x;   // 0..624
    const int seg  = blockIdx.y;   // 0..2
    const int lane = threadIdx.x;

    const float* A; const float* WT; const float* B;
    if (seg == 0)      { A = des;   WT = WT_des;   B = b_des;   }
    else if (seg == 1) { A = tweet; WT = WT_text;  B = b_text;  }
    else               { A = pre_x; WT = WT_tweet; B = b_tweet; }

    const int m  = lane & 15;
    const int kh = lane >> 4;

    v8f c0[5], c1[5];
    #pragma unroll
    for (int t = 0; t < 5; ++t) { c0[t] = (v8f){}; c1[t] = (v8f){}; }

    const float* brow0 = WT + (size_t)m        * LMDIM;  // B col m
    const float* brow1 = WT + (size_t)(16 + m) * LMDIM;  // B col 16+m

    for (int k = 0; k < LMDIM; k += 4) {
        const int ka = k + 2 * kh;
        const v2f b0 = *(const v2f*)(brow0 + ka);
        const v2f b1 = *(const v2f*)(brow1 + ka);
        #pragma unroll
        for (int t = 0; t < 5; ++t) {
            const int row = (5 * tg + t) * 16 + m;
            const v2f a = *(const v2f*)(A + (size_t)row * LMDIM + ka);
            c0[t] = wmma_f32(a, b0, c0[t]);
            c1[t] = wmma_f32(a, b1, c1[t]);
        }
    }

    const int colbase = 64 + seg * HB;
    const float bb0 = B[m], bb1 = B[16 + m];
    #pragma unroll
    for (int t = 0; t < 5; ++t) {
        const int rbase = (5 * tg + t) * 16 + 8 * kh;
        #pragma unroll
        for (int i = 0; i < 8; ++i) {
            h0[(size_t)(rbase + i) * HDIM + colbase + m]      = leaky(c0[t][i] + bb0);
            h0[(size_t)(rbase + i) * HDIM + colbase + 16 + m] = leaky(c1[t][i] + bb1);
        }
    }
}

// ---------------------------------------------------------------------------
// Generic: out = act(A@Wl + bl (+ X@Wr)); WTl/WTr are transposed [160,160].
// grid (625, 5), block 32.  One wave: 5 row-tiles x 32 cols.
// ---------------------------------------------------------------------------
__global__ __launch_bounds__(32) void lin_kernel(
    const float* __restrict__ A,   const float* __restrict__ WTl,
    const float* __restrict__ bl,  const float* __restrict__ X,
    const float* __restrict__ WTr, float* __restrict__ out, int apply_act)
{
    const int tg   = blockIdx.x;   // 0..624
    const int cg   = blockIdx.y;   // 0..4 -> cols cg*32..
    const int lane = threadIdx.x;
    const int col0 = cg * 32;
    const int m  = lane & 15;
    const int kh = lane >> 4;

    v8f c0[5], c1[5];
    #pragma unroll
    for (int t = 0; t < 5; ++t) { c0[t] = (v8f){}; c1[t] = (v8f){}; }

    {
        const float* brow0 = WTl + (size_t)(col0 + m)      * HDIM;
        const float* brow1 = WTl + (size_t)(col0 + 16 + m) * HDIM;
        for (int k = 0; k < HDIM; k += 4) {
            const int ka = k + 2 * kh;
            const v2f b0 = *(const v2f*)(brow0 + ka);
            const v2f b1 = *(const v2f*)(brow1 + ka);
            #pragma unroll
            for (int t = 0; t < 5; ++t) {
                const int row = (5 * tg + t) * 16 + m;
                const v2f a = *(const v2f*)(A + (size_t)row * HDIM + ka);
                c0[t] = wmma_f32(a, b0, c0[t]);
                c1[t] = wmma_f32(a, b1, c1[t]);
            }
        }
    }
    if (X != nullptr) {
        const float* brow0 = WTr + (size_t)(col0 + m)      * HDIM;
        const float* brow1 = WTr + (size_t)(col0 + 16 + m) * HDIM;
        for (int k = 0; k < HDIM; k += 4) {
            const int ka = k + 2 * kh;
            const v2f b0 = *(const v2f*)(brow0 + ka);
            const v2f b1 = *(const v2f*)(brow1 + ka);
            #pragma unroll
            for (int t = 0; t < 5; ++t) {
                const int row = (5 * tg + t) * 16 + m;
                const v2f a = *(const v2f*)(X + (size_t)row * HDIM + ka);
                c0[t] = wmma_f32(a, b0, c0[t]);
                c1[t] = wmma_f32(a, b1, c1[t]);
            }
        }
    }

    const float bb0 = bl[col0 + m], bb1 = bl[col0 + 16 + m];
    #pragma unroll
    for (int t = 0; t < 5; ++t) {
        const int rbase = (5 * tg + t) * 16 + 8 * kh;
        #pragma unroll
        for (int i = 0; i < 8; ++i) {
            float v0 = c0[t][i] + bb0;
            float v1 = c1[t][i] + bb1;
            if (apply_act) { v0 = leaky(v0); v1 = leaky(v1); }
            out[(size_t)(rbase + i) * HDIM + col0 + m]      = v0;
            out[(size_t)(rbase + i) * HDIM + col0 + 16 + m] = v1;
        }
    }
}

// ---------------------------------------------------------------------------
// Edge scatter-add (one wave per edge) + degree count, then mean divide
// ---------------------------------------------------------------------------
__global__ void scatter_kernel(const float* __restrict__ h,
                               const int* __restrict__ src,
                               const int* __restrict__ dst,
                               float* __restrict__ acc,
                               float* __restrict__ deg, long long E)
{
    const long long tid  = (long long)blockIdx.x * blockDim.x + threadIdx.x;
    const long long e    = tid >> 5;
    const int       lane = (int)(tid & 31);
    if (e >= E) return;
    const int s = src[e];
    const int d = dst[e];
    const float* hs = h + (size_t)s * HDIM;
    float* ad = acc + (size_t)d * HDIM;
    #pragma unroll
    for (int j = 0; j < 5; ++j)
        atomicAdd(&ad[lane + 32 * j], hs[lane + 32 * j]);
    if (lane == 0) atomicAdd(&deg[d], 1.0f);
}

__global__ void divide_kernel(float* __restrict__ acc, const float* __restrict__ deg)
{
    const long long tid = (long long)blockIdx.x * blockDim.x + threadIdx.x;
    if (tid >= (long long)NNODES * HDIM) return;
    const int row = (int)(tid / HDIM);
    float d = deg[row];
    d = d > 1.0f ? d : 1.0f;
    acc[tid] = acc[tid] / d;
}

// ---------------------------------------------------------------------------
// em = leaky(h @ W_o1[160,80] + b_o1); WT_o1 transposed [80,160].
// grid (3125, 5), block 32; each wave = 16 rows x 16 cols.
// ---------------------------------------------------------------------------
__global__ __launch_bounds__(32) void head_em_kernel(
    const float* __restrict__ h, const float* __restrict__ WT,
    const float* __restrict__ b, float* __restrict__ em)
{
    const int tile = blockIdx.x;
    const int cg   = blockIdx.y;        // cols cg*16..cg*16+15
    const int lane = threadIdx.x;
    const int row0 = tile * 16;
    const int col0 = cg * 16;
    const int m  = lane & 15;
    const int kh = lane >> 4;

    v8f c0 = {};
    const float* brow = WT + (size_t)(col0 + m) * HDIM;
    for (int k = 0; k < HDIM; k += 4) {
        const int ka = k + 2 * kh;
        const v2f a = *(const v2f*)(h + (size_t)(row0 + m) * HDIM + ka);
        const v2f b0 = *(const v2f*)(brow + ka);
        c0 = wmma_f32(a, b0, c0);
    }
    const int rbase = row0 + 8 * kh;
    const float bb = b[col0 + m];
    #pragma unroll
    for (int i = 0; i < 8; ++i)
        em[(size_t)(rbase + i) * 80 + col0 + m] = leaky(c0[i] + bb);
}

__global__ void head_out_kernel(const float* __restrict__ em,
                                const float* __restrict__ W,
                                const float* __restrict__ b,
                                float* __restrict__ out)
{
    const int tid = blockIdx.x * blockDim.x + threadIdx.x;
    if (tid >= NNODES * 2) return;
    const int row = tid >> 1;
    const int col = tid & 1;
    float s = b[col];
    const float* e = em + (size_t)row * 80;
    #pragma unroll 8
    for (int k = 0; k < 80; ++k) s += e[k] * W[k * 2 + col];
    out[tid] = s;
}

// ---------------------------------------------------------------------------
extern "C" void kernel_launch(void* const* d_in, const int* in_sizes, int n_in,
                              void* d_out, int out_size, void* d_ws, size_t ws_size,
                              hipStream_t stream)
{
    (void)n_in; (void)out_size; (void)ws_size;

    // inputs, setup_inputs() dict order
    const float* pre_x    = (const float*)d_in[0];
    const int*   eidx     = (const int*)d_in[2];
    const float* num_prop = (const float*)d_in[4];
    const float* num_cat  = (const float*)d_in[5];
    const float* des      = (const float*)d_in[6];
    const float* tweet    = (const float*)d_in[7];
    // params, dict insertion order
    const float* W_tweet = (const float*)d_in[8];
    const float* b_tweet = (const float*)d_in[9];
    const float* W_prop  = (const float*)d_in[10];
    const float* b_prop  = (const float*)d_in[11];
    const float* W_cat   = (const float*)d_in[12];
    const float* b_cat   = (const float*)d_in[13];
    const float* W_des   = (const float*)d_in[14];
    const float* b_des   = (const float*)d_in[15];
    const float* W_text  = (const float*)d_in[16];
    const float* b_text  = (const float*)d_in[17];
    const float* W_in    = (const float*)d_in[18];
    const float* b_in    = (const float*)d_in[19];
    const float* W1_l    = (const float*)d_in[20];
    const float* b1_l    = (const float*)d_in[21];
    const float* W1_r    = (const float*)d_in[22];
    const float* W2_l    = (const float*)d_in[23];
    const float* b2_l    = (const float*)d_in[24];
    const float* W2_r    = (const float*)d_in[25];
    const float* W_o1    = (const float*)d_in[26];
    const float* b_o1    = (const float*)d_in[27];
    const float* W_o2    = (const float*)d_in[28];
    const float* b_o2    = (const float*)d_in[29];

    const long long E = in_sizes[2] / 2;
    const int* src = eidx;
    const int* dst = eidx + E;

    // workspace layout
    float* bufA = (float*)d_ws;
    float* bufB = bufA + (size_t)NNODES * HDIM;
    float* bufC = bufB + (size_t)NNODES * HDIM;
    float* deg  = bufC + (size_t)NNODES * HDIM;
    float* wt   = deg  + NNODES;
    float* WT_des   = wt;                 // [32,768]
    float* WT_text  = WT_des   + 32 * LMDIM;
    float* WT_tweet = WT_text  + 32 * LMDIM;
    float* WT_in    = WT_tweet + 32 * LMDIM;   // [160,160] each
    float* WT1_l    = WT_in    + HDIM * HDIM;
    float* WT1_r    = WT1_l    + HDIM * HDIM;
    float* WT2_l    = WT1_r    + HDIM * HDIM;
    float* WT2_r    = WT2_l    + HDIM * HDIM;
    float* WT_o1    = WT2_r    + HDIM * HDIM;  // [80,160]

    float* out_reg = (float*)d_out;                 // [N,2]
    float* em_reg  = out_reg + (size_t)NNODES * 2;  // [N,80]

    // 0) transpose weights (tiny)
    const int t768 = (LMDIM * 32 + 255) / 256;
    const int t160 = (HDIM * HDIM + 255) / 256;
    transpose_kernel<<<t768, 256, 0, stream>>>(W_des,   WT_des,   LMDIM, 32);
    transpose_kernel<<<t768, 256, 0, stream>>>(W_text,  WT_text,  LMDIM, 32);
    transpose_kernel<<<t768, 256, 0, stream>>>(W_tweet, WT_tweet, LMDIM, 32);
    transpose_kernel<<<t160, 256, 0, stream>>>(W_in,  WT_in,  HDIM, HDIM);
    transpose_kernel<<<t160, 256, 0, stream>>>(W1_l,  WT1_l,  HDIM, HDIM);
    transpose_kernel<<<t160, 256, 0, stream>>>(W1_r,  WT1_r,  HDIM, HDIM);
    transpose_kernel<<<t160, 256, 0, stream>>>(W2_l,  WT2_l,  HDIM, HDIM);
    transpose_kernel<<<t160, 256, 0, stream>>>(W2_r,  WT2_r,  HDIM, HDIM);
    transpose_kernel<<<(HDIM * 80 + 255) / 256, 256, 0, stream>>>(W_o1, WT_o1, HDIM, 80);

    const int scatterBlocks = (int)((E * 32 + 255) / 256);
    const int divBlocks     = (NNODES * HDIM + 255) / 256;

    // 1) feature projections -> bufA (h0)
    proj_small_kernel<<<(NNODES * 64 + 255) / 256, 256, 0, stream>>>(
        num_prop, num_cat, W_prop, b_prop, W_cat, b_cat, bufA);
    proj_big_kernel<<<dim3(NTG, 3), 32, 0, stream>>>(
        des, tweet, pre_x, WT_des, b_des, WT_text, b_text, WT_tweet, b_tweet, bufA);

    // 2) h1 = leaky(h0 @ W_in + b_in) -> bufB
    lin_kernel<<<dim3(NTG, 5), 32, 0, stream>>>(bufA, WT_in, b_in, nullptr, nullptr, bufB, 1);

    // 3) SAGE layer 1
    hipMemsetAsync(bufC, 0, (size_t)NNODES * HDIM * sizeof(float), stream);
    hipMemsetAsync(deg,  0, (size_t)NNODES * sizeof(float), stream);
    scatter_kernel<<<scatterBlocks, 256, 0, stream>>>(bufB, src, dst, bufC, deg, E);
    divide_kernel<<<divBlocks, 256, 0, stream>>>(bufC, deg);
    lin_kernel<<<dim3(NTG, 5), 32, 0, stream>>>(bufC, WT1_l, b1_l, bufB, WT1_r, bufA, 0);

    // 4) SAGE layer 2
    hipMemsetAsync(bufC, 0, (size_t)NNODES * HDIM * sizeof(float), stream);
    hipMemsetAsync(deg,  0, (size_t)NNODES * sizeof(float), stream);
    scatter_kernel<<<scatterBlocks, 256, 0, stream>>>(bufA, src, dst, bufC, deg, E);
    divide_kernel<<<divBlocks, 256, 0, stream>>>(bufC, deg);
    lin_kernel<<<dim3(NTG, 5), 32, 0, stream>>>(bufC, WT2_l, b2_l, bufA, WT2_r, bufB, 0);

    // 5) head
    head_em_kernel<<<dim3(NTILES, 5), 32, 0, stream>>>(bufB, WT_o1, b_o1, em_reg);
    head_out_kernel<<<(NNODES * 2 + 255) / 256, 256, 0, stream>>>(em_reg, W_o2, b_o2, out_reg);
}